// HeteroGNN_49194555408762
// MI455X (gfx1250) — compile-verified
//
#include <hip/hip_runtime.h>
#include <math.h>

// ---------------------------------------------------------------------------
// HeteroGNN forward for MI455X (gfx1250, wave32).
// - Dense GEMMs: V_WMMA_F32_16X16X4_F32 (exact fp32, matches reference).
// - A-tile staging: GLOBAL_LOAD_ASYNC_TO_LDS_B128 (ASYNCcnt path) via inline
//   asm, s_wait_asynccnt before the block barrier.
// - Edge aggregation (the real bottleneck, ~9 GB of gather/scatter traffic):
//   wave-per-edge float4 global_atomic_add_f32.
// ---------------------------------------------------------------------------

#define EPSV 1e-5f

typedef float v2f __attribute__((ext_vector_type(2)));
typedef float v8f __attribute__((ext_vector_type(8)));

// C[N,HO] (+)= rowscale(A[N,HK]) @ W[HO,HK]^T + bias, where rowscale is the
// fused segment-mean divide 1/max(rowcnt,1) applied in the epilogue.
// grid.x = ceil(N/16); blockDim = 32*(HO/16) (one wave per 16-wide N tile).
template <int HK, int HO>
__global__ __launch_bounds__(32 * (HO / 16)) void k_gemm_wmma_f32(
    const float* __restrict__ A, const float* __restrict__ W,
    const float* __restrict__ bias, const float* __restrict__ rowcnt,
    float* __restrict__ C, int N, int accumulate) {
  // +4 float pad: row stride 132 -> conflict-free column reads, keeps 16B align
  __shared__ __align__(16) float As[16][HK + 4];
  const int m0 = blockIdx.x * 16;
  const int tid = threadIdx.x;
  const int NT = 32 * (HO / 16);

  // Async copy of the 16 x HK A tile straight into LDS (no VGPR round-trip).
  // Rows past N are clamped to N-1: their garbage only feeds C rows that are
  // never stored. Uniform trip count -> EXEC all ones for every issue.
  {
    const unsigned ldsbase = (unsigned)(size_t)(&As[0][0]);
    const int rowBytes = HK * 4;
    for (int i = tid * 16; i < 16 * rowBytes; i += NT * 16) {
      const int r = i / rowBytes;
      const int cb = i % rowBytes;
      int row = m0 + r;
      if (row >= N) row = N - 1;
      const unsigned lds = ldsbase + (unsigned)(r * (HK + 4) * 4 + cb);
      const unsigned long long ga =
          (unsigned long long)(size_t)((const char*)A + (size_t)row * rowBytes + cb);
      asm volatile("global_load_async_to_lds_b128 %0, %1, off"
                   :: "v"(lds), "v"(ga) : "memory");
    }
    asm volatile("s_wait_asynccnt 0x0" ::: "memory");
  }
  __syncthreads();

  const int lane = tid & 31;
  const int wv = tid >> 5;   // which 16-col N tile this wave owns
  const int l15 = lane & 15;
  const int hi = lane >> 4;  // 0: K=k0..k0+1, 1: K=k0+2..k0+3
  const int n0 = wv * 16;

  v8f acc = {0.f, 0.f, 0.f, 0.f, 0.f, 0.f, 0.f, 0.f};
  const float* wrow = W + (size_t)(n0 + l15) * HK + 2 * hi;
  const float* arow = &As[l15][2 * hi];

#pragma unroll
  for (int k0 = 0; k0 < HK; k0 += 4) {
    // A fragment 16x4: lane l15 holds row m0+l15, cols k0+2*hi, k0+2*hi+1
    const float2 av = *(const float2*)(arow + k0);
    // B fragment 4x16 = W^T tile: lane l15 holds col n0+l15 of W rows
    const float2 wvv = *(const float2*)(wrow + k0);
    v2f a; a.x = av.x; a.y = av.y;
    v2f b; b.x = wvv.x; b.y = wvv.y;
    acc = __builtin_amdgcn_wmma_f32_16x16x4_f32(
        /*neg_a=*/false, a, /*neg_b=*/false, b,
        /*c_mod=*/(short)0, acc, /*reuse_a=*/false, /*reuse_b=*/false);
  }

  // C/D layout: VGPR i -> (M = m0+i, lanes 0-15) / (M = m0+8+i, lanes 16-31),
  // N = n0 + l15.  Segment-mean divide applied here per output row.
  const float bv = bias ? bias[n0 + l15] : 0.0f;
#pragma unroll
  for (int i = 0; i < 8; ++i) {
    const int row = m0 + i + 8 * hi;
    if (row < N) {
      const size_t off = (size_t)row * HO + n0 + l15;
      float o = acc[i];
      if (rowcnt) o *= 1.0f / fmaxf(rowcnt[row], 1.0f);
      o += bv;
      if (accumulate) o += C[off];
      C[off] = o;
    }
  }
}

// One wave per edge: gather 128 floats of x_src[src[e]] (float4/lane), scatter
// atomic-add into sum[dst[e]], count edges per dst.
__global__ void k_scatter_sum(const float* __restrict__ xsrc,
                              const int* __restrict__ esrc,
                              const int* __restrict__ edst,
                              float* __restrict__ sum, float* __restrict__ cnt,
                              int E) {
  const int w = (int)((blockIdx.x * blockDim.x + threadIdx.x) >> 5);
  const int lane = threadIdx.x & 31;
  if (w >= E) return;
  const int s = esrc[w];
  const int d = edst[w];
  const float4 v = *(const float4*)(xsrc + (size_t)s * 128 + lane * 4);
  float* o = sum + (size_t)d * 128 + lane * 4;
  atomicAdd(o + 0, v.x);
  atomicAdd(o + 1, v.y);
  atomicAdd(o + 2, v.z);
  atomicAdd(o + 3, v.w);
  if (lane == 0) atomicAdd(cnt + d, 1.0f);
}

// y = relu( (x*scale - rm)*rsqrt(rv+eps)*g + b ) (+ res)
__global__ void k_bn_relu_res(const float* __restrict__ x,
                              const float* __restrict__ g,
                              const float* __restrict__ b,
                              const float* __restrict__ rm,
                              const float* __restrict__ rv,
                              const float* __restrict__ res,
                              float* __restrict__ y, int total, float scale) {
  const int i = blockIdx.x * blockDim.x + threadIdx.x;
  if (i >= total) return;
  const int c = i & 127;
  float v = x[i] * scale;
  v = (v - rm[c]) * rsqrtf(rv[c] + EPSV) * g[c] + b[c];
  v = fmaxf(v, 0.0f);
  if (res) v += res[i];
  y[i] = v;
}

extern "C" void kernel_launch(void* const* d_in, const int* in_sizes, int n_in,
                              void* d_out, int out_size, void* d_ws,
                              size_t ws_size, hipStream_t stream) {
  (void)n_in; (void)out_size; (void)ws_size;
  const int H = 128;

  // ---- top-level inputs (setup_inputs insertion order) ----
  const float* xin[3]; int N[3];  // 0=drug 1=gene 2=dis
  for (int t = 0; t < 3; ++t) {
    xin[t] = (const float*)d_in[t];
    N[t] = in_sizes[t] / H;
  }
  // relations in d_in order: dg, gd, gs, sg, ds, sd
  const int rel_src[6] = {0, 1, 1, 2, 0, 2};
  const int rel_dst[6] = {1, 0, 2, 1, 2, 0};
  const int* esrc[6]; const int* edst[6]; int E[6];
  for (int r = 0; r < 6; ++r) {
    const int* e = (const int*)d_in[3 + r];
    E[r] = in_sizes[3 + r] / 2;
    esrc[r] = e;
    edst[r] = e + E[r];
  }

  // ---- params: flattened as a JAX pytree (dict keys sorted) ----
  // top: bn_in, final, in, layers
  // bn dicts: b, g, rm, rv ; node types sorted: dis(2), drug(0), gene(1)
  // conv dicts: Wl, Wr, bl ; relations sorted: dg, ds, gd, gs, sd, sg
  int p = 9;
  const int sT[3] = {2, 0, 1};
  const int sR[6] = {0, 4, 1, 2, 5, 3};  // sorted rel -> index in d_in rel order

  const float *bnin_b[3], *bnin_g[3], *bnin_rm[3], *bnin_rv[3];
  for (int k = 0; k < 3; ++k) {
    const int t = sT[k];
    bnin_b[t] = (const float*)d_in[p + 0];
    bnin_g[t] = (const float*)d_in[p + 1];
    bnin_rm[t] = (const float*)d_in[p + 2];
    bnin_rv[t] = (const float*)d_in[p + 3];
    p += 4;
  }
  const float* finW = (const float*)d_in[p + 0];
  const float* finB = (const float*)d_in[p + 1];
  p += 2;
  const float *inW[3], *inB[3];
  for (int k = 0; k < 3; ++k) {
    const int t = sT[k];
    inW[t] = (const float*)d_in[p + 0];
    inB[t] = (const float*)d_in[p + 1];
    p += 2;
  }
  const float *lbn_b[3][3], *lbn_g[3][3], *lbn_rm[3][3], *lbn_rv[3][3];
  const float *Wl[3][6], *Wr[3][6], *bl[3][6];
  for (int l = 0; l < 3; ++l) {
    for (int k = 0; k < 3; ++k) {
      const int t = sT[k];
      lbn_b[l][t] = (const float*)d_in[p + 0];
      lbn_g[l][t] = (const float*)d_in[p + 1];
      lbn_rm[l][t] = (const float*)d_in[p + 2];
      lbn_rv[l][t] = (const float*)d_in[p + 3];
      p += 4;
    }
    for (int k = 0; k < 6; ++k) {
      const int r = sR[k];
      Wl[l][r] = (const float*)d_in[p + 0];
      Wr[l][r] = (const float*)d_in[p + 1];
      bl[l][r] = (const float*)d_in[p + 2];
      p += 3;
    }
  }

  // ---- workspace carve-up ----
  char* ws = (char*)d_ws;
  size_t off = 0;
  auto alloc = [&](size_t bytes) -> float* {
    float* r = (float*)(ws + off);
    off = (off + bytes + 255) & ~(size_t)255;
    return r;
  };
  float *xA[3], *xB[3], *agg[3];
  for (int t = 0; t < 3; ++t) xA[t] = alloc((size_t)N[t] * H * 4);
  for (int t = 0; t < 3; ++t) xB[t] = alloc((size_t)N[t] * H * 4);
  for (int t = 0; t < 3; ++t) agg[t] = alloc((size_t)N[t] * H * 4);
  int maxN = N[0];
  if (N[1] > maxN) maxN = N[1];
  if (N[2] > maxN) maxN = N[2];
  float* mean = alloc((size_t)maxN * H * 4);
  float* cnt = alloc((size_t)maxN * 4);

  auto gemm128 = [&](const float* A, const float* W, const float* bias,
                     const float* rc, float* C, int n, int acc) {
    k_gemm_wmma_f32<128, 128>
        <<<dim3((n + 15) / 16), dim3(256), 0, stream>>>(A, W, bias, rc, C, n, acc);
  };
  auto bnrelu = [&](const float* x, const float* g, const float* b,
                    const float* rm, const float* rv, const float* res,
                    float* y, int n, float scale) {
    const int tot = n * H;
    k_bn_relu_res<<<dim3((tot + 255) / 256), dim3(256), 0, stream>>>(
        x, g, b, rm, rv, res, y, tot, scale);
  };

  // ---- input projection + BN + ReLU ----
  for (int t = 0; t < 3; ++t) {
    gemm128(xin[t], inW[t], inB[t], nullptr, agg[t], N[t], 0);
    bnrelu(agg[t], bnin_g[t], bnin_b[t], bnin_rm[t], bnin_rv[t], nullptr,
           xA[t], N[t], 1.0f);
  }

  // ---- 3 hetero-SAGE layers ----
  for (int l = 0; l < 3; ++l) {
    for (int t = 0; t < 3; ++t)
      hipMemsetAsync(agg[t], 0, (size_t)N[t] * H * 4, stream);
    for (int r = 0; r < 6; ++r) {
      const int s = rel_src[r], d = rel_dst[r];
      hipMemsetAsync(mean, 0, (size_t)N[d] * H * 4, stream);
      hipMemsetAsync(cnt, 0, (size_t)N[d] * 4, stream);
      k_scatter_sum<<<dim3((E[r] + 7) / 8), dim3(256), 0, stream>>>(
          xA[s], esrc[r], edst[r], mean, cnt, E[r]);
      // (sum)@Wl^T, divide-by-count fused into the GEMM epilogue, + bl
      gemm128(mean, Wl[l][r], bl[l][r], cnt, agg[d], N[d], 1);
      // + x_dst@Wr^T
      gemm128(xA[d], Wr[l][r], nullptr, nullptr, agg[d], N[d], 1);
    }
    // HeteroConv mean over 2 relations per dst type (scale=0.5) + BN+ReLU+res
    for (int t = 0; t < 3; ++t)
      bnrelu(agg[t], lbn_g[l][t], lbn_b[l][t], lbn_rm[l][t], lbn_rv[l][t],
             xA[t], xB[t], N[t], 0.5f);
    for (int t = 0; t < 3; ++t) {
      float* tmp = xA[t]; xA[t] = xB[t]; xB[t] = tmp;
    }
  }

  // ---- final 128 -> 64 head, outputs concatenated drug|gene|dis ----
  float* outp = (float*)d_out;
  size_t o = 0;
  for (int t = 0; t < 3; ++t) {
    k_gemm_wmma_f32<128, 64><<<dim3((N[t] + 15) / 16), dim3(128), 0, stream>>>(
        xA[t], finW, finB, nullptr, outp + o, N[t], 0);
    o += (size_t)N[t] * 64;
  }
}